// BodyInterpenetration_66614942761469
// MI455X (gfx1250) — compile-verified
//
#include <hip/hip_runtime.h>

// Problem constants from the reference
#define B_DIM   16
#define NV_DIM  10475
#define F_DIM   20908
#define MAXC    8
#define P_DIM   (F_DIM * MAXC)     // 167264 pairs per batch
#define LIN_MAX 1000.0f

typedef __attribute__((ext_vector_type(2))) float v2f;
typedef __attribute__((ext_vector_type(8))) float v8f;

constexpr int BLOCK = 256;                 // 8 waves (wave32)
constexpr int ITER  = 8;                   // pairs per thread
constexpr int PAIRS_PER_BLOCK = BLOCK * ITER;

__global__ void zero_out_kernel(float* __restrict__ out) {
    if (threadIdx.x < B_DIM) out[threadIdx.x] = 0.0f;
}

__global__ void pen_kernel(const float* __restrict__ v,      // (B, NV, 3)
                           const int*   __restrict__ faces,  // (F, 3)
                           const int*   __restrict__ cidx,   // (B, P, 2)
                           float*       __restrict__ out)    // (B,)
{
    const int b   = blockIdx.y;
    const int tid = threadIdx.x;

    const float* __restrict__ vb = v    + (size_t)b * (size_t)(NV_DIM * 3);
    const int*   __restrict__ cb = cidx + (size_t)b * (size_t)(2 * P_DIM);

    float val = 0.0f;   // per-thread partial sum; stays 0 for masked/out-of-range pairs

    const int pbase = blockIdx.x * PAIRS_PER_BLOCK;

#pragma unroll
    for (int it = 0; it < ITER; ++it) {
        const int p = pbase + it * BLOCK + tid;   // coalesced across the block
        if (p >= P_DIM) continue;

        const int i0 = cb[2 * p + 0];
        const int i1 = cb[2 * p + 1];
        if ((i0 | i1) < 0) continue;              // mask: either index negative

        // Intruder triangle (face i0): 3 vertex gathers
        const int a0 = faces[3 * i0 + 0], a1 = faces[3 * i0 + 1], a2 = faces[3 * i0 + 2];
        const float p0x = vb[3*a0+0], p0y = vb[3*a0+1], p0z = vb[3*a0+2];
        const float p1x = vb[3*a1+0], p1y = vb[3*a1+1], p1z = vb[3*a1+2];
        const float p2x = vb[3*a2+0], p2y = vb[3*a2+1], p2z = vb[3*a2+2];

        // Receiver triangle (face i1)
        const int r0 = faces[3 * i1 + 0], r1 = faces[3 * i1 + 1], r2 = faces[3 * i1 + 2];
        const float q0x = vb[3*r0+0], q0y = vb[3*r0+1], q0z = vb[3*r0+2];
        const float q1x = vb[3*r1+0], q1y = vb[3*r1+1], q1z = vb[3*r1+2];
        const float q2x = vb[3*r2+0], q2y = vb[3*r2+1], q2z = vb[3*r2+2];

        // Centroid of receiver
        const float cx = (q0x + q1x + q2x) * (1.0f / 3.0f);
        const float cy = (q0y + q1y + q2y) * (1.0f / 3.0f);
        const float cz = (q0z + q1z + q2z) * (1.0f / 3.0f);

        // Unit normal of receiver
        const float e1x = q1x - q0x, e1y = q1y - q0y, e1z = q1z - q0z;
        const float e2x = q2x - q0x, e2y = q2y - q0y, e2z = q2z - q0z;
        float nx = e1y * e2z - e1z * e2y;
        float ny = e1z * e2x - e1x * e2z;
        float nz = e1x * e2y - e1y * e2x;
        const float inv = 1.0f / (sqrtf(nx * nx + ny * ny + nz * nz) + 1e-12f);
        nx *= inv; ny *= inv; nz *= inv;

        // Signed depth of the 3 intruder vertices along -n, clamped to [0, 1000]
        float t0 = -((p0x - cx) * nx + (p0y - cy) * ny + (p0z - cz) * nz);
        float t1 = -((p1x - cx) * nx + (p1y - cy) * ny + (p1z - cz) * nz);
        float t2 = -((p2x - cx) * nx + (p2y - cy) * ny + (p2z - cz) * nz);
        t0 = fminf(fmaxf(t0, 0.0f), LIN_MAX);
        t1 = fminf(fmaxf(t1, 0.0f), LIN_MAX);
        t2 = fminf(fmaxf(t2, 0.0f), LIN_MAX);

        val += t0 * t0 + t1 * t1 + t2 * t2;
    }

    // ---- Wave-level reduction via V_WMMA_F32_16X16X4_F32 ----
    // A (16x4 f32, 2 VGPRs): VGPR0 = per-lane partial (K=0 from lanes 0-15,
    // K=2 from lanes 16-31), VGPR1 = 0. B = all-ones (exact x1.0 multiply).
    // D[m][n] = val(lane m) + val(lane m+16) for every n; column N=0 lives in
    // lanes 0 (rows 0-7) and 16 (rows 8-15) of the 8 accumulator VGPRs.
    v2f amat;
    amat[0] = val;
    amat[1] = 0.0f;
    v2f bones;
    bones[0] = 1.0f;
    bones[1] = 1.0f;
    v8f acc = {};
    acc = __builtin_amdgcn_wmma_f32_16x16x4_f32(
        /*neg_a=*/false, amat, /*neg_b=*/false, bones,
        /*c_mod=*/(short)0, acc, /*reuse_a=*/false, /*reuse_b=*/false);

    __shared__ float red[2 * (BLOCK / 32)];   // 2 partials per wave

    const int lane = tid & 31;
    const int wave = tid >> 5;
    if (lane == 0 || lane == 16) {
        float s = acc[0] + acc[1] + acc[2] + acc[3] +
                  acc[4] + acc[5] + acc[6] + acc[7];
        red[wave * 2 + (lane >> 4)] = s;
    }
    __syncthreads();

    if (tid == 0) {
        float s = 0.0f;
#pragma unroll
        for (int i = 0; i < 2 * (BLOCK / 32); ++i) s += red[i];
        atomicAdd(&out[b], s);   // one global_atomic_add_f32 per block
    }
}

extern "C" void kernel_launch(void* const* d_in, const int* in_sizes, int n_in,
                              void* d_out, int out_size, void* d_ws, size_t ws_size,
                              hipStream_t stream) {
    (void)in_sizes; (void)n_in; (void)out_size; (void)d_ws; (void)ws_size;

    const float* v     = (const float*)d_in[0];   // (B, NV, 3) f32
    const int*   faces = (const int*)  d_in[1];   // (F, 3)    i32
    const int*   cidx  = (const int*)  d_in[2];   // (B, P, 2) i32
    float*       out   = (float*)d_out;           // (B,) f32

    // Atomically-accumulated output must start at zero every call.
    zero_out_kernel<<<1, 64, 0, stream>>>(out);

    dim3 grid((P_DIM + PAIRS_PER_BLOCK - 1) / PAIRS_PER_BLOCK, B_DIM, 1);
    pen_kernel<<<grid, BLOCK, 0, stream>>>(v, faces, cidx, out);
}